// TaskAwareMoE_24318104830186
// MI455X (gfx1250) — compile-verified
//
#include <hip/hip_runtime.h>
#include <hip/hip_bf16.h>

// Problem constants (from reference): B=8, N=2048, D=768, E=8, K=2, T=4
#define D_DIM 768
#define E_NUM 8
#define NTOK  16384      // B*N
#define NPB   2048       // N (tokens per batch)
#define NT48  (D_DIM/16) // 48 N-tiles

typedef __attribute__((ext_vector_type(2))) float v2f;
typedef __attribute__((ext_vector_type(8))) float v8f;

__device__ __forceinline__ float gelu_exact(float x) {
    // 0.5*x*(1+erf(x/sqrt(2))) -- matches approximate=False
    return 0.5f * x * (1.0f + erff(x * 0.70710678118654752440f));
}

// ---------------------------------------------------------------------------
// Kernel 0: per-batch gate bias = gate_b + task_embed[task_id] @ gate_W[D:,:]
// ---------------------------------------------------------------------------
__global__ __launch_bounds__(64)
void TaskAwareMoE_taskbias_kernel(const int* __restrict__ task_ids,
                                  const float* __restrict__ task_embed,
                                  const float* __restrict__ gate_W,
                                  const float* __restrict__ gate_b,
                                  float* __restrict__ taskbias) {
    const int t = threadIdx.x;          // 64 threads: (b,e)
    const int b = t >> 3;
    const int e = t & 7;
    const int tb = task_ids[b];
    float acc = gate_b[e];
    const float* tv = task_embed + (size_t)tb * D_DIM;
    for (int d = 0; d < D_DIM; ++d)
        acc += tv[d] * gate_W[(size_t)(D_DIM + d) * E_NUM + e];
    taskbias[b * E_NUM + e] = acc;
}

// ---------------------------------------------------------------------------
// Kernel 1: per-token gating: logits -> top-2 softmax -> gates[8], omega
// One wave per token, lanes stride over D.
// ---------------------------------------------------------------------------
__global__ __launch_bounds__(256)
void TaskAwareMoE_gate_kernel(const float* __restrict__ tokens,
                              const float* __restrict__ gate_W,
                              const float* __restrict__ taskbias,
                              float* __restrict__ gates,
                              float* __restrict__ omega) {
    const int wave = threadIdx.x >> 5;
    const int lane = threadIdx.x & 31;
    const int m = blockIdx.x * 8 + wave;       // token index
    const int b = m / NPB;

    float p[E_NUM];
#pragma unroll
    for (int e = 0; e < E_NUM; ++e) p[e] = 0.0f;

    const float* x = tokens + (size_t)m * D_DIM;
    for (int d = lane; d < D_DIM; d += 32) {
        const float xv = x[d];
        const float* wrow = gate_W + (size_t)d * E_NUM;
#pragma unroll
        for (int e = 0; e < E_NUM; ++e) p[e] += xv * wrow[e];
    }
    // wave32 butterfly reduce (all lanes end with the total)
#pragma unroll
    for (int off = 16; off > 0; off >>= 1) {
#pragma unroll
        for (int e = 0; e < E_NUM; ++e) p[e] += __shfl_xor(p[e], off, 32);
    }

    if (lane == 0) {
        float l[E_NUM];
#pragma unroll
        for (int e = 0; e < E_NUM; ++e) l[e] = p[e] + taskbias[b * E_NUM + e];
        // top-2 (strict > keeps lowest index on ties, like lax.top_k)
        int i1 = 0;
#pragma unroll
        for (int e = 1; e < E_NUM; ++e) if (l[e] > l[i1]) i1 = e;
        int i2 = (i1 == 0) ? 1 : 0;
#pragma unroll
        for (int e = 0; e < E_NUM; ++e) if (e != i1 && l[e] > l[i2]) i2 = e;
        const float v1 = l[i1], v2 = l[i2];
        const float g1 = 1.0f / (1.0f + expf(v2 - v1));  // softmax over {v1,v2}
        const float g2 = 1.0f - g1;
        float go[E_NUM];
#pragma unroll
        for (int e = 0; e < E_NUM; ++e) go[e] = 0.0f;
        go[i1] = g1;
        go[i2] = g2;
#pragma unroll
        for (int e = 0; e < E_NUM; ++e) gates[(size_t)m * E_NUM + e] = go[e];
        omega[m] = 1.0f - g1;              // 1 - max(gates)
    }
}

// ---------------------------------------------------------------------------
// Kernel 2: fused 9-way GEMM (8 experts + universal) with f32 WMMA.
// Grid: (NTOK/128) x 48 n-tiles. Block: 256 threads = 8 waves.
// Wave w: 16 tokens (mTile*128 + w*16 ..) x 16 features (nTile*16 ..),
// 9 accumulators held simultaneously so A fragments load once per K-step.
// Weight panels staged in LDS, shared by all 8 waves.
// ---------------------------------------------------------------------------
__global__ __launch_bounds__(256)
void TaskAwareMoE_moe_wmma_kernel(const float* __restrict__ tokens,
                                  const float* __restrict__ W_e,
                                  const float* __restrict__ b_e,
                                  const float* __restrict__ W_u,
                                  const float* __restrict__ b_u,
                                  const float* __restrict__ gates,
                                  const float* __restrict__ omega,
                                  float* __restrict__ out) {
    __shared__ float ldsB[9][32][16];      // 18 KB: K-chunk x 16 cols x 9 mats

    const int nTile = blockIdx.x % NT48;
    const int mTile = blockIdx.x / NT48;
    const int nBase = nTile * 16;
    const int tid   = threadIdx.x;
    const int wave  = tid >> 5;
    const int lane  = tid & 31;
    const int laneLo = lane & 15;          // column / M-row selector
    const int laneHi = lane >> 4;          // 0/1: K-pair / +8 row half
    const int mWave = mTile * 128 + wave * 16;

    v8f acc[9];
#pragma unroll
    for (int e = 0; e < 9; ++e) acc[e] = (v8f){};

    // A fragment base: lane<16 -> row laneLo, K=(k,k+1); lane>=16 -> K=(k+2,k+3)
    const float* aBase = tokens + (size_t)(mWave + laneLo) * D_DIM + laneHi * 2;

    for (int k0 = 0; k0 < D_DIM; k0 += 32) {
        __syncthreads();
        // cooperative stage: 9 weight panels, 32(K) x 16(N) each
        for (int idx = tid; idx < 9 * 32 * 16; idx += 256) {   // exactly 18 iters
            const int e   = idx >> 9;
            const int rem = idx & 511;
            const int k   = rem >> 4;
            const int n   = rem & 15;
            const float* Wp = (e < 8) ? (W_e + (size_t)e * D_DIM * D_DIM) : W_u;
            ldsB[e][k][n] = Wp[(size_t)(k0 + k) * D_DIM + nBase + n];
        }
        __syncthreads();

#pragma unroll
        for (int kk = 0; kk < 32; kk += 4) {
            const v2f a = *reinterpret_cast<const v2f*>(aBase + k0 + kk);
#pragma unroll
            for (int e = 0; e < 9; ++e) {
                v2f bfrag;
                bfrag.x = ldsB[e][kk + laneHi * 2 + 0][laneLo];
                bfrag.y = ldsB[e][kk + laneHi * 2 + 1][laneLo];
                acc[e] = __builtin_amdgcn_wmma_f32_16x16x4_f32(
                    false, a, false, bfrag, (short)0, acc[e], false, false);
            }
        }
    }

    // epilogue: bias + exact GELU + gate/omega combine
    float bias[9];
#pragma unroll
    for (int e = 0; e < E_NUM; ++e) bias[e] = b_e[(size_t)e * D_DIM + nBase + laneLo];
    bias[8] = b_u[nBase + laneLo];

#pragma unroll
    for (int i = 0; i < 8; ++i) {
        const int m = mWave + i + laneHi * 8;      // C layout: row = i (+8 for hi lanes)
        const float* g = gates + (size_t)m * E_NUM;
        const float om = omega[m];
        float v = om * gelu_exact(acc[8][i] + bias[8]);
#pragma unroll
        for (int e = 0; e < E_NUM; ++e)
            v += g[e] * gelu_exact(acc[e][i] + bias[e]);
        out[(size_t)m * D_DIM + nBase + laneLo] = v;
    }
}

// ---------------------------------------------------------------------------
extern "C" void kernel_launch(void* const* d_in, const int* in_sizes, int n_in,
                              void* d_out, int out_size, void* d_ws, size_t ws_size,
                              hipStream_t stream) {
    const float* tokens     = (const float*)d_in[0];
    const int*   task_ids   = (const int*)d_in[1];
    const float* task_embed = (const float*)d_in[2];
    const float* gate_W     = (const float*)d_in[3];
    const float* gate_b     = (const float*)d_in[4];
    const float* W_e        = (const float*)d_in[5];
    const float* b_e        = (const float*)d_in[6];
    const float* W_u        = (const float*)d_in[7];
    const float* b_u        = (const float*)d_in[8];
    float* out = (float*)d_out;

    // workspace layout (floats): taskbias[64] | gates[16384*8] | omega[16384]
    float* taskbias = (float*)d_ws;
    float* gates    = taskbias + 64;
    float* omega    = gates + (size_t)NTOK * E_NUM;

    TaskAwareMoE_taskbias_kernel<<<1, 64, 0, stream>>>(
        task_ids, task_embed, gate_W, gate_b, taskbias);

    TaskAwareMoE_gate_kernel<<<NTOK / 8, 256, 0, stream>>>(
        tokens, gate_W, taskbias, gates, omega);

    TaskAwareMoE_moe_wmma_kernel<<<(NTOK / 128) * NT48, 256, 0, stream>>>(
        tokens, W_e, b_e, W_u, b_u, gates, omega, out);
}